// NSSD_71227737637389
// MI455X (gfx1250) — compile-verified
//
#include <hip/hip_runtime.h>
#include <math.h>

// ---------------------------------------------------------------------------
// Problem constants (from reference): N=32768, feat=2048, S=8, d=256.
// Output: d_out[0] = loss ; d_out[1..2048] = new cluster centers [8][256].
// Workspace (f32): [0]=loss1 acc, [1]=loss2 acc, [2..2049]=gn-sum acc,
//                  [2050..6145] = normalized cluster centers padded to 16x256.
// ---------------------------------------------------------------------------

typedef float v2f __attribute__((ext_vector_type(2)));
typedef float v8f __attribute__((ext_vector_type(8)));

#define STRIPES     8
#define DDIM        256
#define FEAT        2048
#define WAVES_PB    8
#define THREADS_PB  256
#define GRID_MAIN   256
#define WS_CCN_OFF  2050

// ---------------- Kernel 0: zero accumulators + normalize cluster centers ---
__global__ __launch_bounds__(THREADS_PB) void nssd_prep(
    const float* __restrict__ cc, float* __restrict__ ws) {
  const int tid  = threadIdx.x;
  const int wave = tid >> 5;
  const int lane = tid & 31;

  // zero loss + gn-sum accumulators (must re-init every launch)
  for (int i = tid; i < WS_CCN_OFF; i += THREADS_PB) ws[i] = 0.0f;

  float* ccn = ws + WS_CCN_OFF;
  // zero the pad rows 8..15 of the B matrix
  for (int i = tid; i < STRIPES * DDIM; i += THREADS_PB)
    ccn[STRIPES * DDIM + i] = 0.0f;

  // wave w normalizes center row w (8 elements per lane)
  const float* row = cc + wave * DDIM;
  float x[8];
  float ss = 0.0f;
#pragma unroll
  for (int j = 0; j < 8; ++j) {
    x[j] = row[lane + 32 * j];
    ss += x[j] * x[j];
  }
#pragma unroll
  for (int m = 16; m >= 1; m >>= 1) ss += __shfl_xor(ss, m, 32);
  const float sc = 1.0f / (sqrtf(ss) + 1e-12f);
  float* orow = ccn + wave * DDIM;
#pragma unroll
  for (int j = 0; j < 8; ++j) orow[lane + 32 * j] = x[j] * sc;
}

// ---------------- Kernel 1: main WMMA pipeline ------------------------------
__global__ __launch_bounds__(THREADS_PB) void nssd_main(
    const float* __restrict__ gfeat, float* __restrict__ ws, int N) {
  __shared__ float lds_sum[STRIPES * DDIM];

  const int tid  = threadIdx.x;
  const int wave = tid >> 5;
  const int lane = tid & 31;

  for (int i = tid; i < STRIPES * DDIM; i += THREADS_PB) lds_sum[i] = 0.0f;
  __syncthreads();

  const float* ccn  = ws + WS_CCN_OFF;
  float*       gsum = ws + 2;

  // WMMA f32 16x16x4 fragment geometry (wave32):
  //  A: lane L<16 holds row L, K = {4k, 4k+1}; lane L>=16 holds row L-16,
  //     K = {4k+2, 4k+3}.  B fragment of A^T has identical lane data.
  const int r      = lane & 15;   // A row / D col index 0..15
  const int half   = lane >> 4;   // K sub-offset selector
  const int stripe = r & 7;       // stripe within sample
  const int col    = r;           // output column this lane owns

  const float* brow = ccn + (size_t)col * DDIM + 2 * half;  // rows 8..15 = 0

  float l1p = 0.0f, l2p = 0.0f;
  const int npair = N >> 1;

  for (int pair = blockIdx.x * WAVES_PB + wave; pair < npair;
       pair += GRID_MAIN * WAVES_PB) {
    const size_t n = (size_t)2 * pair + (r >> 3);  // rows 0-7: sample a, 8-15: b
    const float* arow = gfeat + n * FEAT + stripe * DDIM + 2 * half;

    // warm the next tile for this wave (global_prefetch_b8)
    __builtin_prefetch(arow + (size_t)2 * GRID_MAIN * WAVES_PB * FEAT, 0, 0);

    v8f c  = {0.f, 0.f, 0.f, 0.f, 0.f, 0.f, 0.f, 0.f};  // Gram accum
    v8f dm = {0.f, 0.f, 0.f, 0.f, 0.f, 0.f, 0.f, 0.f};  // center-dot accum
#pragma unroll
    for (int k = 0; k < 64; ++k) {
      v2f a = *(const v2f*)(arow + 4 * k);
      v2f b = *(const v2f*)(brow + 4 * k);
      c  = __builtin_amdgcn_wmma_f32_16x16x4_f32(false, a, false, a,
                                                 (short)0, c, false, false);
      dm = __builtin_amdgcn_wmma_f32_16x16x4_f32(false, a, false, b,
                                                 (short)0, dm, false, false);
    }

    // ---- extract inverse norms from Gram diagonal --------------------------
    // Diag(row rr) lives at vgpr rr&7, lane rr (rr<8) or lane rr+16 (rr>=8).
    float myc = c[0];
#pragma unroll
    for (int j = 1; j < 8; ++j) myc = ((lane & 7) == j) ? c[j] : myc;
    const float invd = 1.0f / (sqrtf(fmaxf(myc, 0.0f)) + 1e-12f);

    float invRow[8];  // inv norm of output row (v + 8*half)
#pragma unroll
    for (int v = 0; v < 8; ++v) invRow[v] = __shfl(invd, v + 24 * half, 32);
    // inv norm of row 'col' (== A-row r); used for loss1 cols and gn scaling
    const float invC = __shfl(invd, (col < 8) ? col : col + 16, 32);

    // ---- loss1: sum |C_norm - I| over both per-sample 8x8 blocks -----------
#pragma unroll
    for (int v = 0; v < 8; ++v) {
      const int  row   = v + 8 * half;
      const bool valid = ((row < 8) == (col < 8));
      if (valid) {
        const float val = c[v] * invRow[v] * invC;
        const float tgt = (row == col) ? 1.0f : 0.0f;
        l1p += fabsf(val - tgt);
      }
    }

    // ---- loss2: softplus(diag - min_t(D + I)) per (sample, stripe) ---------
#pragma unroll
    for (int v = 0; v < 8; ++v) {
      const float dot = dm[v] * invRow[v];
      float Dst = fmaxf((1.0f - dot) * 0.5f, 1e-12f);
      float x   = (col < 8) ? (Dst + ((col == v) ? 1.0f : 0.0f)) : 3.0e38f;
#pragma unroll
      for (int m = 1; m <= 4; m <<= 1) x = fminf(x, __shfl_xor(x, m, 32));
      if (col == v) l2p += log1pf(expf(Dst - x));  // diag lane of valid group
    }

    // ---- gn-sum: re-read tile (WGP$/L2 hot) scaled by inv norm -> LDS ------
    const int base = stripe * DDIM + 2 * half;
#pragma unroll
    for (int k = 0; k < 64; ++k) {
      v2f a = *(const v2f*)(arow + 4 * k);
      unsafeAtomicAdd(&lds_sum[base + 4 * k + 0], a.x * invC);
      unsafeAtomicAdd(&lds_sum[base + 4 * k + 1], a.y * invC);
    }
  }

  // ---- wave-reduce loss partials, flush to global accumulators -------------
#pragma unroll
  for (int m = 1; m <= 16; m <<= 1) {
    l1p += __shfl_xor(l1p, m, 32);
    l2p += __shfl_xor(l2p, m, 32);
  }
  if (lane == 0) {
    unsafeAtomicAdd(&ws[0], l1p);
    unsafeAtomicAdd(&ws[1], l2p);
  }
  __syncthreads();
  for (int i = tid; i < STRIPES * DDIM; i += THREADS_PB)
    unsafeAtomicAdd(&gsum[i], lds_sum[i]);
}

// ---------------- Kernel 2: finalize ----------------------------------------
__global__ __launch_bounds__(THREADS_PB) void nssd_fin(
    const float* __restrict__ cc, const float* __restrict__ ws,
    float* __restrict__ out, int N) {
  const int tid = threadIdx.x;
  if (tid == 0) {
    const float l1 = ws[0] / ((float)N * 64.0f);  // /(N*S*S)
    const float l2 = ws[1] / ((float)N * 8.0f);   // mean over N*S
    out[0] = l1 + l2;
  }
  for (int i = tid; i < STRIPES * DDIM; i += THREADS_PB)
    out[1 + i] = 0.9f * cc[i] + 0.1f * (ws[2 + i] / (float)N);
}

// ---------------- Host launch ----------------------------------------------
extern "C" void kernel_launch(void* const* d_in, const int* in_sizes, int n_in,
                              void* d_out, int out_size, void* d_ws,
                              size_t ws_size, hipStream_t stream) {
  const float* gfeat = (const float*)d_in[0];
  const float* cc    = (const float*)d_in[1];
  float*       out   = (float*)d_out;
  float*       ws    = (float*)d_ws;
  const int    N     = in_sizes[0] / FEAT;

  nssd_prep<<<1, THREADS_PB, 0, stream>>>(cc, ws);
  nssd_main<<<GRID_MAIN, THREADS_PB, 0, stream>>>(gfeat, ws, N);
  nssd_fin<<<1, THREADS_PB, 0, stream>>>(cc, ws, out, N);
}